// Classify_Extractor_80066780332737
// MI455X (gfx1250) — compile-verified
//
#include <hip/hip_runtime.h>

typedef __attribute__((ext_vector_type(2))) float v2f;
typedef __attribute__((ext_vector_type(4))) float v4f;
typedef __attribute__((ext_vector_type(8))) float v8f;

#define BB 512
#define LL 2048
#define TT 16
// START_TAG = 0, STOP_TAG = 1

__device__ __forceinline__ float fexp(float x) {
    return __builtin_amdgcn_exp2f(x * 1.44269504088896340736f);
}
__device__ __forceinline__ float flog(float x) {
    return __builtin_amdgcn_logf(x) * 0.69314718055994530942f;
}

// ---------------------------------------------------------------------------
// Forward algorithm (partition function) via f32 WMMA.
// One wave (32 threads) per group of 16 batch elements; alpha state lives in
// the WMMA C/D layout: reg r on lane l holds tag j = r + 8*(l>=16), batch
// column b = l&15.  Per step:  D = exp(trans) (16x16)  x  P (16x16),
// P[i,b] = exp(alpha[i,b] - m[b]);  alpha_new = m + log(D) + feat (masked).
// ---------------------------------------------------------------------------
__global__ void __launch_bounds__(32)
crf_forward_kernel(const float* __restrict__ feats,
                   const int*   __restrict__ mask,
                   const float* __restrict__ trans,
                   float*       __restrict__ fwd_out)
{
    __shared__ float P[TT * 16];          // p[tag][batch-col] staging (1 KB)

    const int lane = threadIdx.x;         // 0..31 (wave32)
    const int hi   = lane >> 4;           // lane half
    const int bl   = lane & 15;           // batch column / matrix row index
    const int b    = blockIdx.x * 16 + bl;

    // A operand: exp(trans[j,i]) in 16x4 f32 A-layout, 4 K-chunks.
    // VGPR0: lanes0-15 K=4c, lanes16-31 K=4c+2; VGPR1: K=4c+1 / K=4c+3.
    v2f A[4];
#pragma unroll
    for (int c = 0; c < 4; ++c) {
        const int k0 = 4 * c + 2 * hi;
        A[c].x = fexp(trans[bl * TT + k0 + 0]);
        A[c].y = fexp(trans[bl * TT + k0 + 1]);
    }

    // STOP-tag transition row for this lane's 8 tags (j = r + 8*hi)
    float tstop[8];
#pragma unroll
    for (int r = 0; r < 8; ++r) tstop[r] = trans[1 * TT + (r + 8 * hi)];

    // init alpha: -10000 everywhere, 0 at START_TAG (j==0)
    float alpha[8];
#pragma unroll
    for (int r = 0; r < 8; ++r)
        alpha[r] = ((r + 8 * hi) == 0) ? 0.0f : -10000.0f;

    const size_t fbase = (size_t)b * (LL * TT) + (size_t)(8 * hi);
    const int*   mrow  = mask + (size_t)b * LL;

    for (int t = 0; t < LL; ++t) {
        // emissions: 8 consecutive f32, feats[b, t, 8*hi .. 8*hi+7]
        const float* fp = feats + fbase + (size_t)t * TT;
        __builtin_prefetch(fp + 4 * TT, 0, 0);        // global_prefetch_b8
        const v4f f0 = *(const v4f*)(fp);
        const v4f f1 = *(const v4f*)(fp + 4);
        const int mt = mrow[t];

        // column max m[b] over 16 tags (8 regs + cross-half exchange)
        float m = alpha[0];
#pragma unroll
        for (int r = 1; r < 8; ++r) m = fmaxf(m, alpha[r]);
        m = fmaxf(m, __shfl_xor(m, 16, 32));

        // P[i,b] = exp(alpha[i,b] - m[b]) staged to LDS
#pragma unroll
        for (int r = 0; r < 8; ++r)
            P[(r + 8 * hi) * 16 + bl] = fexp(alpha[r] - m);
        __syncthreads();

        // D = E x P : four independent 16x16x4 f32 WMMAs (break dep chain)
        v8f acc0 = {}, acc1 = {}, acc2 = {}, acc3 = {};
        {
            v2f Bv;
            int tg;
            tg = 0 + 2 * hi;                       // chunk 0: K rows 0..3
            Bv.x = P[tg * 16 + bl]; Bv.y = P[(tg + 1) * 16 + bl];
            acc0 = __builtin_amdgcn_wmma_f32_16x16x4_f32(false, A[0], false, Bv,
                                                         (short)0, acc0, false, false);
            tg = 4 + 2 * hi;                       // chunk 1: K rows 4..7
            Bv.x = P[tg * 16 + bl]; Bv.y = P[(tg + 1) * 16 + bl];
            acc1 = __builtin_amdgcn_wmma_f32_16x16x4_f32(false, A[1], false, Bv,
                                                         (short)0, acc1, false, false);
            tg = 8 + 2 * hi;                       // chunk 2: K rows 8..11
            Bv.x = P[tg * 16 + bl]; Bv.y = P[(tg + 1) * 16 + bl];
            acc2 = __builtin_amdgcn_wmma_f32_16x16x4_f32(false, A[2], false, Bv,
                                                         (short)0, acc2, false, false);
            tg = 12 + 2 * hi;                      // chunk 3: K rows 12..15
            Bv.x = P[tg * 16 + bl]; Bv.y = P[(tg + 1) * 16 + bl];
            acc3 = __builtin_amdgcn_wmma_f32_16x16x4_f32(false, A[3], false, Bv,
                                                         (short)0, acc3, false, false);
        }
        __syncthreads();
        const v8f acc = (acc0 + acc1) + (acc2 + acc3);

        const float femit[8] = { f0[0], f0[1], f0[2], f0[3],
                                 f1[0], f1[1], f1[2], f1[3] };
#pragma unroll
        for (int r = 0; r < 8; ++r) {
            const float an = m + flog(acc[r]) + femit[r];
            alpha[r] = mt ? an : alpha[r];
        }
    }

    // terminal: logsumexp_j(alpha[j,b] + trans[STOP, j])
    float term[8];
    float m2 = -3.4e38f;
#pragma unroll
    for (int r = 0; r < 8; ++r) { term[r] = alpha[r] + tstop[r]; m2 = fmaxf(m2, term[r]); }
    m2 = fmaxf(m2, __shfl_xor(m2, 16, 32));
    float s = 0.0f;
#pragma unroll
    for (int r = 0; r < 8; ++r) s += fexp(term[r] - m2);
    s += __shfl_xor(s, 16, 32);
    if (hi == 0) fwd_out[b] = m2 + flog(s);
}

// ---------------------------------------------------------------------------
// Gold path score: one wave per batch element, lanes strided over L,
// deterministic shfl-tree reduction (mask is a contiguous prefix).
// ---------------------------------------------------------------------------
__global__ void __launch_bounds__(256)
crf_gold_kernel(const float* __restrict__ feats,
                const int*   __restrict__ mask,
                const int*   __restrict__ tgt,
                const float* __restrict__ trans,
                float*       __restrict__ gold_out)
{
    const int wid  = (blockIdx.x * blockDim.x + threadIdx.x) >> 5;  // batch
    const int lane = threadIdx.x & 31;
    if (wid >= BB) return;

    const int*   trow = tgt  + (size_t)wid * LL;
    const int*   mrow = mask + (size_t)wid * LL;
    const float* frow = feats + (size_t)wid * LL * TT;

    float s = 0.0f;
    int   cnt = 0;
    for (int l = lane; l < LL; l += 32) {
        const int cur  = trow[l];
        const int prev = (l == 0) ? 0 : trow[l - 1];   // START_TAG = 0
        if (mrow[l]) {
            s   += trans[cur * TT + prev] + frow[(size_t)l * TT + cur];
            cnt += 1;
        }
    }
#pragma unroll
    for (int off = 16; off > 0; off >>= 1) {
        s   += __shfl_xor(s, off, 32);
        cnt += __shfl_xor(cnt, off, 32);
    }
    if (lane == 0) {
        const int last_tag = trow[cnt - 1];            // lengths >= L/2 >= 1
        gold_out[wid] = s + trans[1 * TT + last_tag];  // + trans[STOP, last]
    }
}

// ---------------------------------------------------------------------------
// Final deterministic reduction: mean_b(fwd[b] - gold[b])
// ---------------------------------------------------------------------------
__global__ void __launch_bounds__(32)
crf_reduce_kernel(const float* __restrict__ fwd,
                  const float* __restrict__ gold,
                  float*       __restrict__ out)
{
    const int lane = threadIdx.x;
    float s = 0.0f;
#pragma unroll
    for (int i = 0; i < BB / 32; ++i) {
        const int b = lane * (BB / 32) + i;
        s += fwd[b] - gold[b];
    }
#pragma unroll
    for (int off = 16; off > 0; off >>= 1) s += __shfl_xor(s, off, 32);
    if (lane == 0) out[0] = s * (1.0f / (float)BB);
}

extern "C" void kernel_launch(void* const* d_in, const int* in_sizes, int n_in,
                              void* d_out, int out_size, void* d_ws, size_t ws_size,
                              hipStream_t stream) {
    (void)in_sizes; (void)n_in; (void)out_size; (void)ws_size;
    const float* feats = (const float*)d_in[0];   // [B, L, T] f32
    const int*   mask  = (const int*)  d_in[1];   // [B, L] i32
    const int*   tgt   = (const int*)  d_in[2];   // [B, L] i32
    const float* trans = (const float*)d_in[3];   // [T, T] f32
    float* out = (float*)d_out;

    float* ws_fwd  = (float*)d_ws;        // [512]
    float* ws_gold = ws_fwd + BB;         // [512]

    crf_forward_kernel<<<BB / 16, 32, 0, stream>>>(feats, mask, trans, ws_fwd);
    crf_gold_kernel<<<(BB * 32 + 255) / 256, 256, 0, stream>>>(feats, mask, tgt, trans, ws_gold);
    crf_reduce_kernel<<<1, 32, 0, stream>>>(ws_fwd, ws_gold, out);
}